// Model_65524021068103
// MI455X (gfx1250) — compile-verified
//
#include <hip/hip_runtime.h>
#include <math.h>

// Problem constants (from reference)
#define SEQ_LEN 336
#define PRED_LEN 96
#define WIN     12
#define RES     25
#define WASH    10
#define IN_SEG  28          // SEQ_LEN / WIN
#define PRED_SEG 8
#define RO_IN   18          // IN_SEG - WASH
#define HID     32
#define NCH     21
#define BATCH   4096
#define KPAD    28          // reservoir dim padded 25 -> 28 (multiple of 4 for WMMA K)
#define JTOT    (RO_IN * KPAD)   // 504 = collapsed readout K dimension
#define NOUT    96

typedef __attribute__((ext_vector_type(2))) float v2f;
typedef __attribute__((ext_vector_type(8))) float v8f;

// FP32 WMMA: D(16x16) = A(16x4) * B(4x16) + C   -> v_wmma_f32_16x16x4_f32
static __device__ __forceinline__ v8f wmma_f32(v2f a, v2f b, v8f c) {
    return __builtin_amdgcn_wmma_f32_16x16x4_f32(
        /*neg_a=*/false, a, /*neg_b=*/false, b,
        /*c_mod=*/(short)0, c, /*reuse_a=*/false, /*reuse_b=*/false);
}

static __device__ __forceinline__ float fast_tanh(float x) {
#if defined(__has_builtin)
#if __has_builtin(__builtin_amdgcn_tanhf)
    return __builtin_amdgcn_tanhf(x);
#else
    return tanhf(x);
#endif
#else
    return tanhf(x);
#endif
}

// ---------------------------------------------------------------------------
// Kernel 1: collapse the linear readout (fc1 -> fc2 -> proj) into a single
// K=504, N=96 weight matrix per channel, stored transposed [o][j] so the
// main kernel can fetch B fragments with 64-bit loads. Also folds all biases.
// ---------------------------------------------------------------------------
__global__ void __launch_bounds__(256)
prep_kernel(const float* __restrict__ fc1_w, const float* __restrict__ fc1_b,
            const float* __restrict__ fc2_w, const float* __restrict__ fc2_b,
            const float* __restrict__ proj_w, const float* __restrict__ proj_b,
            float* __restrict__ WpT, float* __restrict__ bias) {
    const int c = blockIdx.x;
    const int tid = threadIdx.x;

    __shared__ float G[PRED_SEG][RO_IN];   // G[p][l] = sum_h fc2_w[p,h]*fc1_w[h,l]
    __shared__ float S[WIN];               // S[w]    = sum_r proj_w[w,r]
    __shared__ float beta[PRED_SEG];       // beta[p] = fc1_b . fc2_w[p] + fc2_b[p]

    for (int idx = tid; idx < PRED_SEG * RO_IN; idx += blockDim.x) {
        int p = idx / RO_IN, l = idx % RO_IN;
        float acc = 0.f;
        for (int h = 0; h < HID; ++h)
            acc += fc2_w[(c * PRED_SEG + p) * HID + h] * fc1_w[(c * HID + h) * RO_IN + l];
        G[p][l] = acc;
    }
    for (int w = tid; w < WIN; w += blockDim.x) {
        float s = 0.f;
        for (int r = 0; r < RES; ++r) s += proj_w[(c * WIN + w) * RES + r];
        S[w] = s;
    }
    for (int p = tid; p < PRED_SEG; p += blockDim.x) {
        float b = fc2_b[c * PRED_SEG + p];
        for (int h = 0; h < HID; ++h)
            b += fc1_b[c * HID + h] * fc2_w[(c * PRED_SEG + p) * HID + h];
        beta[p] = b;
    }
    __syncthreads();

    // WpT[c][o][j]: o = p*12+w (96 outputs), j = l*28+r (K, r>=25 zero-padded)
    float* wc = WpT + (size_t)c * NOUT * JTOT;
    for (int idx = tid; idx < NOUT * JTOT; idx += blockDim.x) {
        int o = idx / JTOT, j = idx % JTOT;
        int p = o / WIN, w = o % WIN;
        int l = j / KPAD, r = j % KPAD;
        float v = 0.f;
        if (r < RES) v = proj_w[(c * WIN + w) * RES + r] * G[p][l];
        wc[idx] = v;
    }
    for (int o = tid; o < NOUT; o += blockDim.x) {
        int p = o / WIN, w = o % WIN;
        bias[c * NOUT + o] = beta[p] * S[w] + proj_b[c * WIN + w];
    }
}

// ---------------------------------------------------------------------------
// Kernel 2: fused ESN. One wave32 = 16 batch rows x 1 channel.
//   per step t: u_t via 6 WMMAs (K=12), recurrence via 14 WMMAs (K=28),
//   tanh, LDS D->A transpose, and (t>=WASH) 42 WMMAs accumulating the
//   16x96 output tile against the collapsed readout matrix.
// WMMA fragment conventions (wave32, f32 16x16x4):
//   A(16x4):  lane L, vgpr v -> (M = L%16, K = 2*(L/16) + v)
//   B(4x16):  lane L, vgpr v -> (K = 2*(L/16) + v, N = L%16)
//   C/D:      lane L, vgpr e -> (M = e + 8*(L/16), N = L%16)
// ---------------------------------------------------------------------------
__global__ void __launch_bounds__(32)
esn_kernel(const float* __restrict__ x, const float* __restrict__ W_in,
           const float* __restrict__ W_res, const float* __restrict__ WpT,
           const float* __restrict__ bias, float* __restrict__ out) {
    const int c    = blockIdx.y;
    const int m0   = blockIdx.x * 16;
    const int lane = threadIdx.x;
    const int lo   = lane & 15;
    const int hi   = lane >> 4;

    __shared__ float hbuf[16 * KPAD];   // 16x28 transpose scratch (1.75 KB)

    // B fragments of W_in^T: K = window idx (12 -> 3 chunks), N = reservoir idx
    v2f Bin[3][2];
#pragma unroll
    for (int kc = 0; kc < 3; ++kc)
#pragma unroll
        for (int nt = 0; nt < 2; ++nt) {
            int n = nt * 16 + lo;
#pragma unroll
            for (int v = 0; v < 2; ++v) {
                int k = kc * 4 + hi * 2 + v;          // always < 12
                Bin[kc][nt][v] = (n < RES) ? W_in[(c * RES + n) * WIN + k] : 0.f;
            }
        }

    // B fragments of W_res^T: K = source res (25 -> 7 chunks), N = dest res
    v2f Bres[7][2];
#pragma unroll
    for (int kc = 0; kc < 7; ++kc)
#pragma unroll
        for (int nt = 0; nt < 2; ++nt) {
            int n = nt * 16 + lo;
#pragma unroll
            for (int v = 0; v < 2; ++v) {
                int k = kc * 4 + hi * 2 + v;
                Bres[kc][nt][v] = (k < RES && n < RES) ? W_res[(c * RES + n) * RES + k] : 0.f;
            }
        }

    // recurrent state as A fragments, h0 = 0 (K padded to 28 stays 0 throughout)
    v2f hA[7];
#pragma unroll
    for (int kc = 0; kc < 7; ++kc) hA[kc] = {};

    // 16x96 output accumulator (6 N-tiles)
    v8f oacc[6];
#pragma unroll
    for (int nt = 0; nt < 6; ++nt) oacc[nt] = {};

    const float* xb  = x + (size_t)(m0 + lo) * SEQ_LEN * NCH + c;
    const float* wct = WpT + (size_t)c * NOUT * JTOT;

    for (int t = 0; t < IN_SEG; ++t) {
        // A fragments of this segment's input window (16 x 12)
        v2f axs[3];
#pragma unroll
        for (int kc = 0; kc < 3; ++kc) {
            int k = kc * 4 + hi * 2;
            axs[kc][0] = xb[(t * WIN + k) * NCH];
            axs[kc][1] = xb[(t * WIN + k + 1) * NCH];
        }

        // pre-activation = xs_t @ W_in^T + h @ W_res^T  (chained accumulation)
        v8f acc[2];
#pragma unroll
        for (int nt = 0; nt < 2; ++nt) {
            acc[nt] = {};
#pragma unroll
            for (int kc = 0; kc < 3; ++kc) acc[nt] = wmma_f32(axs[kc], Bin[kc][nt], acc[nt]);
#pragma unroll
            for (int kc = 0; kc < 7; ++kc) acc[nt] = wmma_f32(hA[kc], Bres[kc][nt], acc[nt]);
        }

        // tanh, then D -> LDS (rows M=0..15, cols N=0..27; cols 25..27 are 0)
#pragma unroll
        for (int nt = 0; nt < 2; ++nt) {
            int n = nt * 16 + lo;
#pragma unroll
            for (int e = 0; e < 8; ++e) {
                float hval = fast_tanh(acc[nt][e]);
                if (n < KPAD) hbuf[(e + 8 * hi) * KPAD + n] = hval;
            }
        }
        __syncthreads();
        // reload as A fragments for the next step / readout (64-bit DS loads)
#pragma unroll
        for (int kc = 0; kc < 7; ++kc)
            hA[kc] = *(const v2f*)&hbuf[lo * KPAD + kc * 4 + hi * 2];
        __syncthreads();

        // fused collapsed readout: oacc += state_t @ W'[rows l*28 .. l*28+27]
        if (t >= WASH) {
            const float* wl = wct + (t - WASH) * KPAD;
#pragma unroll
            for (int nt = 0; nt < 6; ++nt) {
                const float* wn = wl + (size_t)(nt * 16 + lo) * JTOT;  // WpT[o][...]
#pragma unroll
                for (int kc = 0; kc < 7; ++kc) {
                    v2f bw = *(const v2f*)&wn[kc * 4 + hi * 2];
                    oacc[nt] = wmma_f32(hA[kc], bw, oacc[nt]);
                }
            }
        }
    }

    // out[b, o, c] with o = nt*16 + lo, b = m0 + e + 8*hi
#pragma unroll
    for (int nt = 0; nt < 6; ++nt) {
        int o = nt * 16 + lo;
        float bv = bias[c * NOUT + o];
#pragma unroll
        for (int e = 0; e < 8; ++e) {
            int b = m0 + e + 8 * hi;
            out[((size_t)b * NOUT + o) * NCH + c] = oacc[nt][e] + bv;
        }
    }
}

extern "C" void kernel_launch(void* const* d_in, const int* in_sizes, int n_in,
                              void* d_out, int out_size, void* d_ws, size_t ws_size,
                              hipStream_t stream) {
    const float* x      = (const float*)d_in[0];
    const float* W_in   = (const float*)d_in[1];
    const float* W_res  = (const float*)d_in[2];
    const float* fc1_w  = (const float*)d_in[3];
    const float* fc1_b  = (const float*)d_in[4];
    const float* fc2_w  = (const float*)d_in[5];
    const float* fc2_b  = (const float*)d_in[6];
    const float* proj_w = (const float*)d_in[7];
    const float* proj_b = (const float*)d_in[8];
    float* out = (float*)d_out;

    // workspace: collapsed readout weights (transposed) + folded bias (~4.07 MB)
    float* WpT  = (float*)d_ws;                          // 21 * 96 * 504 floats
    float* bias = WpT + (size_t)NCH * NOUT * JTOT;       // 21 * 96 floats

    prep_kernel<<<dim3(NCH), dim3(256), 0, stream>>>(fc1_w, fc1_b, fc2_w, fc2_b,
                                                     proj_w, proj_b, WpT, bias);
    esn_kernel<<<dim3(BATCH / 16, NCH), dim3(32), 0, stream>>>(x, W_in, W_res,
                                                               WpT, bias, out);
}